// SSTInputLayerV2_23502061044031
// MI455X (gfx1250) — compile-verified
//
#include <hip/hip_runtime.h>

// ---------------- static config (mirrors reference) ----------------
#define NVOX    100000
#define CDIM    128
#define WINX    16
#define WINY    16
#define NWX     31
#define NWY     31
#define NWZ     2
#define NWIN    3844            // BATCH(2) * 31*31*2
#define T_MAX   144
#define SLOTS   (NWIN * T_MAX)  // 553,536
#define NBLK    ((NVOX + 255) / 256)   // 391

#define SZ_FEAT ((long long)NWIN * T_MAX * CDIM)   // 70,852,608
#define SZ_MASK ((long long)NWIN * T_MAX)          // 553,536
#define PER_SHIFT (2*SZ_FEAT + SZ_MASK + 3LL*NVOX) // 142,558,752

typedef float v4f __attribute__((ext_vector_type(4)));   // native vector: OK for nontemporal builtins

// ---------------- CDNA5 async-LDS helpers (gfx1250 ISA) ----------------
__device__ __forceinline__ void async_load_b128(unsigned lds_off, unsigned long long gaddr) {
    // copies 16 bytes from global address gaddr to LDS offset lds_off   (ASYNCcnt-tracked, regular temporal:
    // feats row is re-read on the second shift pass -> keep it L2-resident)
    asm volatile("global_load_async_to_lds_b128 %0, %1, off"
                 :: "v"(lds_off), "v"(gaddr) : "memory");
}
__device__ __forceinline__ void async_store_b128_nt(unsigned long long gaddr, unsigned lds_off) {
    // copies 16 bytes from LDS offset lds_off to global address gaddr   (ASYNCcnt-tracked, non-temporal:
    // write-once streaming output, don't evict feats/scratch from the 192MB L2)
    asm volatile("global_store_async_from_lds_b128 %0, %1, off th:TH_STORE_NT"
                 :: "v"(gaddr), "v"(lds_off) : "memory");
}
__device__ __forceinline__ void wait_async0() {
    asm volatile("s_wait_asynccnt 0" ::: "memory");
}
__device__ __forceinline__ void nt_store_v4(float* p, v4f v) {
    __builtin_nontemporal_store(v, (v4f*)p);
}

// ---------------- K0: inverse map init ----------------
__global__ void init_s2v_kernel(int* __restrict__ s2v) {
    int i = blockIdx.x * blockDim.x + threadIdx.x;
    if (i < SLOTS) s2v[i] = -1;                    // re-read by gather -> regular temporal
}

// ---------------- K1: window hash + in-window coords ----------------
__global__ void coords_kernel(const int* __restrict__ coors, int shifted,
                              int* __restrict__ win_v, int* __restrict__ cxy) {
    int i = blockIdx.x * blockDim.x + threadIdx.x;
    if (i >= NVOX) return;
    int b = coors[4*i + 0];
    int z = coors[4*i + 1];                       // always 0, shift_z = 0
    int y = coors[4*i + 2] + (shifted ? (WINY/2) : 0);
    int x = coors[4*i + 3] + (shifted ? (WINX/2) : 0);
    int wx = x >> 4, wy = y >> 4, wz = z;         // WIN = (16,16,1)
    int w = b * (NWX*NWY*NWZ) + wx * (NWY*NWZ) + wy * NWZ + wz;
    win_v[i] = w;
    cxy[i] = (x & 15) | ((y & 15) << 8);
}

// ---------------- K2: stable local rank + per-block window histogram ----------------
__global__ void rank_hist_kernel(const int* __restrict__ win_v,
                                 int* __restrict__ lrank, int* __restrict__ hist) {
    __shared__ int swin[256];
    __shared__ int shist[NWIN];                   // 15,376 B of LDS (of 320KB/WGP)
    int tid = threadIdx.x;
    int gid = blockIdx.x * 256 + tid;
    for (int i = tid; i < NWIN; i += 256) shist[i] = 0;
    int w = (gid < NVOX) ? win_v[gid] : -1;
    swin[tid] = w;
    __syncthreads();
    if (gid < NVOX) {
        int r = 0;
        for (int j = 0; j < tid; ++j) r += (swin[j] == w) ? 1 : 0;  // stable in-block rank
        lrank[gid] = r;
        atomicAdd(&shist[w], 1);                  // ds_add_u32
    }
    __syncthreads();
    long long base = (long long)blockIdx.x * NWIN;
    for (int i = tid; i < NWIN; i += 256) hist[base + i] = shist[i];
}

// ---------------- K3: exclusive prefix across blocks, per window ----------------
__global__ void scan_kernel(int* __restrict__ hist, int* __restrict__ counts) {
    int w = blockIdx.x * blockDim.x + threadIdx.x;
    if (w >= NWIN) return;
    int run = 0;
    for (int b = 0; b < NBLK; ++b) {              // coalesced across windows
        long long idx = (long long)b * NWIN + w;
        int t = hist[idx];
        hist[idx] = run;
        run += t;
    }
    counts[w] = run;
}

// ---------------- K4: inner index, drop level, inverse map ----------------
__global__ void assign_kernel(const int* __restrict__ win_v, const int* __restrict__ lrank,
                              const int* __restrict__ hist, const int* __restrict__ counts,
                              int* __restrict__ s2v, float* __restrict__ lvl_o,
                              float* __restrict__ inner_o, float* __restrict__ win_o) {
    int i = blockIdx.x * 256 + threadIdx.x;
    if (i >= NVOX) return;
    int w = win_v[i];
    int inner = lrank[i] + hist[(long long)(i >> 8) * NWIN + w];
    int n = counts[w];
    int lvl = (n < 30) ? 0 : (n < 60) ? 1 : (n < 100000) ? 2 : -1;  // DROP_INFO
    // small write-once outputs -> non-temporal
    __builtin_nontemporal_store((float)lvl,   lvl_o + i);
    __builtin_nontemporal_store((float)inner, inner_o + i);
    __builtin_nontemporal_store((float)w,     win_o + i);
    if (inner < T_MAX) s2v[w * T_MAX + inner] = i;   // unique per slot -> deterministic
}

// ---------------- K5: one wave32 per padded slot; async-LDS feature copy + pos embed ----------------
__global__ void gather_kernel(const float* __restrict__ feats, const int* __restrict__ s2v,
                              const int* __restrict__ cxy, float* __restrict__ feat3d,
                              float* __restrict__ pos3d, float* __restrict__ maskf) {
    __shared__ float lbuf[8 * CDIM];              // 512 B per wave
    int lane = threadIdx.x & 31;
    int wv   = threadIdx.x >> 5;
    long long slot = (long long)blockIdx.x * 8 + wv;
    if (slot >= SLOTS) return;                    // wave-uniform exit
    int v = s2v[slot];
    if (lane == 0)                                // key_mask: True at padding (streaming)
        __builtin_nontemporal_store((v < 0) ? 1.0f : 0.0f, maskf + slot);
    float* fd = feat3d + slot * CDIM + lane * 4;
    float* pd = pos3d  + slot * CDIM + lane * 4;
    if (v < 0) {
        v4f z = (v4f){0.f, 0.f, 0.f, 0.f};
        nt_store_v4(fd, z);
        nt_store_v4(pd, z);
    } else {
        // launch async copy of this lane's 16B of the feature row through LDS
        unsigned lds_off = (unsigned)(unsigned long long)&lbuf[wv * CDIM + lane * 4];
        unsigned long long src = (unsigned long long)(feats + (long long)v * CDIM + lane * 4);
        async_load_b128(lds_off, src);
        // overlap load latency with pos-embed math:
        // channels c = lane*4 .. lane*4+3 ; lanes 0..15 -> x part, 16..31 -> y part
        int pk = cxy[v];
        float base = (lane < 16) ? (float)(pk & 0xff)        - (WINX * 0.5f)
                                 : (float)((pk >> 8) & 0xff) - (WINY * 0.5f);
        int j0 = (lane & 15) * 2;                 // freq index pair
        float f0 = __powf(10000.0f, (float)j0       * (1.0f / 32.0f));
        float f1 = __powf(10000.0f, (float)(j0 + 1) * (1.0f / 32.0f));
        float a0 = base / f0, a1 = base / f1;
        v4f p = (v4f){sinf(a0), cosf(a0), sinf(a1), cosf(a1)};
        nt_store_v4(pd, p);
        wait_async0();                            // LDS now holds the row
        async_store_b128_nt((unsigned long long)fd, lds_off);
        // S_ENDPGM performs implicit wait-idle; LDS not reused afterwards.
    }
}

extern "C" void kernel_launch(void* const* d_in, const int* in_sizes, int n_in,
                              void* d_out, int out_size, void* d_ws, size_t ws_size,
                              hipStream_t stream) {
    const float* feats = (const float*)d_in[0];   // [NVOX, 128] f32
    const int*   coors = (const int*)d_in[1];     // [NVOX, 4]  i32 (b,z,y,x)
    float* out = (float*)d_out;

    // workspace carve-up (reused across the two shifts; ~9.5 MB)
    int* win_v  = (int*)d_ws;            // NVOX
    int* cxy    = win_v + NVOX;          // NVOX
    int* lrank  = cxy + NVOX;            // NVOX
    int* counts = lrank + NVOX;          // NWIN
    int* s2v    = counts + NWIN;         // SLOTS
    int* hist   = s2v + SLOTS;           // NBLK * NWIN

    for (int s = 0; s < 2; ++s) {
        float* o       = out + (long long)s * PER_SHIFT;
        float* feat3d  = o;
        float* pos3d   = o + SZ_FEAT;
        float* maskf   = o + 2 * SZ_FEAT;
        float* lvl_o   = maskf + SZ_MASK;
        float* inner_o = lvl_o + NVOX;
        float* win_o   = inner_o + NVOX;

        init_s2v_kernel<<<(SLOTS + 255) / 256, 256, 0, stream>>>(s2v);
        coords_kernel<<<(NVOX + 255) / 256, 256, 0, stream>>>(coors, s, win_v, cxy);
        rank_hist_kernel<<<NBLK, 256, 0, stream>>>(win_v, lrank, hist);
        scan_kernel<<<(NWIN + 255) / 256, 256, 0, stream>>>(hist, counts);
        assign_kernel<<<(NVOX + 255) / 256, 256, 0, stream>>>(
            win_v, lrank, hist, counts, s2v, lvl_o, inner_o, win_o);
        gather_kernel<<<(SLOTS + 7) / 8, 256, 0, stream>>>(
            feats, s2v, cxy, feat3d, pos3d, maskf);
    }
}